// AdjacencyControl_81793357185324
// MI455X (gfx1250) — compile-verified
//
#include <hip/hip_runtime.h>

// CDNA5 / gfx1250, wave32. Pipeline:
//   1) mask[i]  = (rank[i] <= 10000)
//   2) Wt2: k-paired transpose of W -> Wt2[(k/2)*2D + 2n + (k&1)] = W[n][k]
//      (64 KB one-time; makes each WMMA B-frag one contiguous b64 in LDS)
//   3) out      = 0
//   4) h = x @ W^T + b : per block, TDM stages the 64 KB tile into LDS
//      (tensor_load_to_lds + s_wait_tensorcnt), then each wave computes a
//      16x128 strip with 8 x V_WMMA_F32_16X16X4_F32 per k-step. x is read
//      exactly once (51 MB); B-frags are single ds_load_b64s.
//   5) per edge: if mask[col] -> out[row] += h[col]  (wave/edge, fp32 atomics,
//      ~90% of edges skipped by the scalar mask test; h/out are L2-resident)

typedef __attribute__((ext_vector_type(2))) float        v2f;
typedef __attribute__((ext_vector_type(8))) float        v8f;
typedef __attribute__((ext_vector_type(4))) unsigned int v4u;
typedef __attribute__((ext_vector_type(8))) int          v8i;
typedef __attribute__((ext_vector_type(4))) int          v4i;

#define K_THRESH 10000
#define DIM 128

__global__ void mask_kernel(const int* __restrict__ rank, int* __restrict__ mask, int N) {
    int i = blockIdx.x * blockDim.x + threadIdx.x;
    if (i < N) mask[i] = (rank[i] <= K_THRESH) ? 1 : 0;
}

// k-paired transpose: element (k, n) of W^T stored at (k/2)*2D + 2n + (k&1),
// so {(k,n), (k+1,n)} are 8 contiguous bytes -> one ds_load_b64 per B-frag.
__global__ void transpose_kernel(const float* __restrict__ W, float* __restrict__ Wt2, int D) {
    int i = blockIdx.x * blockDim.x + threadIdx.x;
    if (i < D * D) {
        int n = i / D, k = i % D;
        Wt2[(k >> 1) * (2 * D) + 2 * n + (k & 1)] = W[n * D + k];
    }
}

__global__ void zero_kernel(float4* __restrict__ p, int n4) {
    int i = blockIdx.x * blockDim.x + threadIdx.x;
    if (i < n4) p[i] = make_float4(0.f, 0.f, 0.f, 0.f);
}

// One wave computes a 16x128 strip of h (all 8 column tiles), so x is read
// exactly once. Wt2 lives in LDS, staged by the Tensor Data Mover.
// A-frag (16x4 f32): lanes 0-15 -> row M=l {K=k0,k0+1}; lanes 16-31 -> {k0+2,k0+3}.
// C/D layout: acc[t][v] = element (M = v + 8*half, N = 16*t + l).
__global__ void __launch_bounds__(256)
gemm_wmma_kernel(const float* __restrict__ x, const float* __restrict__ Wt_g,
                 const float* __restrict__ bias, float* __restrict__ h, int N) {
    __shared__ float Wt[DIM * DIM];          // 64 KB, k-paired layout

    const int lane        = threadIdx.x & 31;
    const int waveInBlock = threadIdx.x >> 5;

    // --- Stage the 64 KB tile into LDS via TDM, wave 0 only ---
    if (waveInBlock == 0) {
        unsigned int       lds_addr = (unsigned int)(size_t)(void*)Wt; // LDS offset = low 32b of generic addr
        unsigned long long ga       = (unsigned long long)(size_t)Wt_g;
        v4u g0;
        g0.x = 1u;                                           // count=1, user descriptor
        g0.y = lds_addr;                                     // lds_addr [63:32]
        g0.z = (unsigned int)(ga & 0xFFFFFFFFu);             // global_addr [95:64]
        g0.w = (unsigned int)((ga >> 32) & 0x1FFFFFFu)       // global_addr [120:96]
               | (2u << 30);                                 // type=2 ("image")
        v8i g1;
        g1[0] = (int)(2u << 16);          // data_size=2 -> 4-byte elements; wg_mask=0
        g1[1] = (int)(128u << 16);        // tensor_dim0 = 128   (bits 79:48)
        g1[2] = (int)(128u << 16);        // tensor_dim1 = 128   (bits 111:80)
        g1[3] = (int)(128u << 16);        // tile_dim0   = 128   (bits 127:112)
        g1[4] = 128;                      // tile_dim1   = 128   (bits 143:128)
        g1[5] = 128;                      // tensor_dim0_stride = 128 (bits 207:160)
        g1[6] = 0;
        g1[7] = 0;
        v4i gz4 = {0, 0, 0, 0};           // 2D tensor: groups 2/3 unused
        v8i gz8 = {0, 0, 0, 0, 0, 0, 0, 0};
        __builtin_amdgcn_tensor_load_to_lds(g0, g1, gz4, gz4, gz8, 0);
        __builtin_amdgcn_s_wait_tensorcnt(0);
    }
    __syncthreads();

    const int rowTile = blockIdx.x * (blockDim.x >> 5) + waveInBlock;
    const int m0      = rowTile << 4;
    if (m0 >= N) return;

    const int half = lane >> 4;
    const int l    = lane & 15;

    v8f acc[8] = {};
    for (int k = 0; k < DIM; k += 4) {
        const int ka = k + 2 * half;          // even: k or k+2
        // A: one contiguous b64 per lane from row m0+l of x
        v2f a = *(const v2f*)(x + (size_t)(m0 + l) * DIM + ka);
        const float* brow = &Wt[(ka >> 1) * (2 * DIM)];
#pragma unroll
        for (int t = 0; t < 8; ++t) {
            // B-frag {(ka,n),(ka+1,n)} for n = 16t+l: one aligned ds_load_b64
            v2f bb = *(const v2f*)(brow + 2 * (t * 16 + l));
            acc[t] = __builtin_amdgcn_wmma_f32_16x16x4_f32(
                false, a, false, bb, (short)0, acc[t], false, false);
        }
    }

#pragma unroll
    for (int t = 0; t < 8; ++t) {
        const float bv = bias[t * 16 + l];     // bias depends on column only
#pragma unroll
        for (int v = 0; v < 8; ++v)
            h[(size_t)(m0 + v + 8 * half) * DIM + t * 16 + l] = acc[t][v] + bv;
    }
}

// One wave per edge. row/col made wave-uniform (readfirstlane) so the mask
// test is a scalar branch; ~90% of edges skip all memory traffic.
__global__ void scatter_kernel(const int* __restrict__ row, const int* __restrict__ col,
                               const int* __restrict__ mask, const float* __restrict__ h,
                               float* __restrict__ out, int E) {
    const int wave = (blockIdx.x * blockDim.x + threadIdx.x) >> 5;
    const int lane = threadIdx.x & 31;
    if (wave >= E) return;

    const int c = __builtin_amdgcn_readfirstlane(col[wave]);
    if (mask[c] == 0) return;                  // zero contribution: skip
    const int r = __builtin_amdgcn_readfirstlane(row[wave]);

    // lane L handles features 4L..4L+3: one b128 gather + 4 fp32 atomics
    const float4 hv = *(const float4*)(h + (size_t)c * DIM + lane * 4);
    float* o = out + (size_t)r * DIM + lane * 4;
    unsafeAtomicAdd(o + 0, hv.x);
    unsafeAtomicAdd(o + 1, hv.y);
    unsafeAtomicAdd(o + 2, hv.z);
    unsafeAtomicAdd(o + 3, hv.w);
}

extern "C" void kernel_launch(void* const* d_in, const int* in_sizes, int n_in,
                              void* d_out, int out_size, void* d_ws, size_t ws_size,
                              hipStream_t stream) {
    const float* x    = (const float*)d_in[0];   // [N, D]
    const float* W    = (const float*)d_in[1];   // [D, D]
    const float* b    = (const float*)d_in[2];   // [D]
    const int*   edge = (const int*)d_in[3];     // [2, E] flat: row then col
    const int*   rank = (const int*)d_in[4];     // [N]

    const int D = in_sizes[2];                   // 128
    const int N = in_sizes[4];
    const int E = in_sizes[3] / 2;
    const int* rowp = edge;
    const int* colp = edge + E;

    // Workspace: h [N*D f32] | Wt2 [D*D f32] | mask [N i32]
    float* h    = (float*)d_ws;
    float* Wt2  = h + (size_t)N * D;
    int*   mask = (int*)(Wt2 + (size_t)D * D);
    float* out  = (float*)d_out;

    mask_kernel<<<(N + 255) / 256, 256, 0, stream>>>(rank, mask, N);
    transpose_kernel<<<(D * D + 255) / 256, 256, 0, stream>>>(W, Wt2, D);

    const int n4 = (N * D) / 4;
    zero_kernel<<<(n4 + 255) / 256, 256, 0, stream>>>((float4*)out, n4);

    const int tiles16 = N / 16;                  // 6250 strips of 16 rows
    gemm_wmma_kernel<<<(tiles16 + 7) / 8, 256, 0, stream>>>(x, Wt2, b, h, N);

    const int sblocks = (E + 7) / 8;             // 8 waves / 256-thread block
    scatter_kernel<<<sblocks, 256, 0, stream>>>(rowp, colp, mask, h, out, E);
}